// BAnet_84980222919288
// MI455X (gfx1250) — compile-verified
//
#include <hip/hip_runtime.h>
#include <hip/hip_bf16.h>
#include <cstdint>

typedef __attribute__((ext_vector_type(2))) float v2f;
typedef __attribute__((ext_vector_type(8))) float v8f;

#define N_NODES 8192
#define NB      4
#define SEQQ    12
#define SS      11      /* S = SEQ-KCNN+1 */
#define HH      11
#define SET     32
#define NCOLS   96      /* 2 streams * 4 batch * 12 seq */
#define KCHUNK  64
#define CH_FLOATS (KCHUNK * NCOLS)   /* 6144 floats = 24 KB */
#define CH_VEC4   (CH_FLOATS / 4)    /* 1536 */

// ---------------------------------------------------------------------------
// CDNA5 async memory->LDS helpers (inline asm; ASYNCcnt-tracked)
// ---------------------------------------------------------------------------
__device__ __forceinline__ void async_ld_b128(uint32_t lds_addr, const void* gaddr) {
  asm volatile("global_load_async_to_lds_b128 %0, %1, off"
               :: "v"(lds_addr), "v"(gaddr) : "memory");
}
__device__ __forceinline__ void wait_async0() {
  asm volatile("s_wait_asynccnt 0x0" ::: "memory");
}

// ---------------------------------------------------------------------------
// 1) Pack occ/prc (B,N,SEQ) into X2 in *WMMA-fragment-swizzled* order:
//    idx = ((kb*6 + t)*64 + lane*2 + j)
//      row = kb*4 + (lane>>4) + 2*j,  col = t*16 + (lane&15)
//      col = stream*48 + b*12 + s
//    So the GEMM's B fragment is a single aligned 8-byte LDS load.
// ---------------------------------------------------------------------------
__global__ __launch_bounds__(256)
void pack_x_k(const float* __restrict__ occ, const float* __restrict__ prc,
              float* __restrict__ Xp) {
  int idx = blockIdx.x * 256 + threadIdx.x;
  if (idx >= N_NODES * NCOLS) return;
  int kb   = idx / 384;
  int rem  = idx % 384;
  int t    = rem / 64;
  int lp   = rem % 64;
  int lane = lp >> 1, j = lp & 1;
  int row  = kb * 4 + (lane >> 4) + 2 * j;
  int col  = t * 16 + (lane & 15);
  int stream = col / 48, bc = col % 48, b = bc / SEQQ, s = bc % SEQQ;
  const float* src = stream ? prc : occ;
  Xp[idx] = src[((size_t)b * N_NODES + row) * SEQQ + s];
}

// ---------------------------------------------------------------------------
// 2) RW[8192][96] = rw_adj(8192x8192) * X(8192x96), fp32 WMMA 16x16x4.
//    128 threads = 4 waves; each wave owns 16 rows x all 96 cols (6 n-tiles).
//    B panel async-copied to double-buffered LDS (ASYNCcnt overlap);
//    A streamed from HBM with global_prefetch.
// ---------------------------------------------------------------------------
__global__ __launch_bounds__(128)
void gemm_rw_k(const float* __restrict__ A, const float* __restrict__ X2,
               float* __restrict__ C) {
  __shared__ __align__(16) float Bs[2][CH_FLOATS];   // 48 KB double buffer
  const int tid  = threadIdx.x;
  const int wave = tid >> 5;
  const int lane = tid & 31;
  const int row0 = blockIdx.x * 64 + wave * 16;
  const int m     = lane & 15;          // A: M index (ISA 16x4 layout)
  const int khalf = (lane >> 4) << 1;   // A: K={0,1} (lanes 0-15) / K={2,3}

  const uint32_t lds0 = (uint32_t)(uintptr_t)(&Bs[0][0]);
  const uint32_t lds1 = (uint32_t)(uintptr_t)(&Bs[1][0]);

  v8f acc[6];
  const v8f vzero = {0.f,0.f,0.f,0.f,0.f,0.f,0.f,0.f};
#pragma unroll
  for (int t = 0; t < 6; ++t) acc[t] = vzero;

  // prologue: async-fill chunk 0 into buffer 0
  {
    const float4* g = (const float4*)X2 + tid;
    uint32_t l = lds0 + (uint32_t)tid * 16u;
    for (int i = tid; i < CH_VEC4; i += 128) {
      async_ld_b128(l, g);
      g += 128;  l += 128u * 16u;
    }
  }
  wait_async0();
  __syncthreads();

  const int NCHUNK = N_NODES / KCHUNK;   // 128
  for (int c = 0; c < NCHUNK; ++c) {
    const int k0 = c * KCHUNK;
    // overlap: issue async fill of next chunk into the other buffer
    if (c + 1 < NCHUNK) {
      const float4* g = (const float4*)(X2 + (size_t)(k0 + KCHUNK) * NCOLS) + tid;
      uint32_t l = (((c + 1) & 1) ? lds1 : lds0) + (uint32_t)tid * 16u;
      for (int i = tid; i < CH_VEC4; i += 128) {
        async_ld_b128(l, g);
        g += 128;  l += 128u * 16u;
      }
    }
    const float* Bcur = (c & 1) ? &Bs[1][0] : &Bs[0][0];
    const float* bfbase = Bcur + lane * 2;     // fragment base for this lane
    const float* arow = A + (size_t)(row0 + m) * N_NODES + k0 + khalf;
    if (k0 + KCHUNK < N_NODES)
      __builtin_prefetch(arow + KCHUNK, 0, 1); // global_prefetch_b8

    for (int ks = 0; ks < KCHUNK; ks += 4) {
      v2f a = *(const v2f*)(arow + ks);        // A fragment (2 K values)
      const int kb = ks >> 2;
#pragma unroll
      for (int t = 0; t < 6; ++t) {
        v2f bf = *(const v2f*)(bfbase + (kb * 6 + t) * 64);  // one ds_load_b64
        acc[t] = __builtin_amdgcn_wmma_f32_16x16x4_f32(
            false, a, false, bf, (short)0, acc[t], false, false);
      }
    }
    wait_async0();       // next-chunk fill complete (overlapped with WMMA)
    __syncthreads();     // safe to overwrite the buffer we just read
  }

  // C fragment: VGPR r -> M = r (lanes 0-15) or 8+r (lanes 16-31), N = lane%16
  const int nn   = (lane < 16) ? lane : lane - 16;
  const int mrow = (lane < 16) ? 0 : 8;
#pragma unroll
  for (int t = 0; t < 6; ++t) {
    const int col = t * 16 + nn;
#pragma unroll
    for (int r = 0; r < 8; ++r)
      C[(size_t)(row0 + mrow + r) * NCOLS + col] = acc[t][r];
  }
}

// ---------------------------------------------------------------------------
// 3) enhance epilogue + causal conv -> h_conv (B,N,11)
// ---------------------------------------------------------------------------
__device__ __forceinline__ float sigm(float x) { return 1.f / (1.f + __expf(-x)); }

__global__ __launch_bounds__(256)
void enhance_conv_k(const float* __restrict__ RW, const float* __restrict__ occ,
                    const float* __restrict__ prc, const float* __restrict__ Wt,
                    const float* __restrict__ bt, const float* __restrict__ Wa,
                    const float* __restrict__ ba, const float* __restrict__ cW,
                    const float* __restrict__ cb, float* __restrict__ hconv) {
  int idx = blockIdx.x * 256 + threadIdx.x;
  if (idx >= NB * N_NODES) return;
  int b = idx >> 13, n = idx & (N_NODES - 1);
  float ro[SEQQ], rp[SEQQ], oe[SEQQ], pe[SEQQ];
#pragma unroll
  for (int s = 0; s < SEQQ; ++s) {
    ro[s] = RW[(size_t)n * NCOLS + b * SEQQ + s];
    rp[s] = RW[(size_t)n * NCOLS + 48 + b * SEQQ + s];
  }
  const size_t base = ((size_t)b * N_NODES + n) * SEQQ;
#pragma unroll
  for (int s = 0; s < SEQQ; ++s) {
    float tro = bt[s], ato = ba[s], trp = bt[s], atp = ba[s];
#pragma unroll
    for (int j = 0; j < SEQQ; ++j) {
      tro += Wt[s * SEQQ + j] * ro[j];  ato += Wa[s * SEQQ + j] * ro[j];
      trp += Wt[s * SEQQ + j] * rp[j];  atp += Wa[s * SEQQ + j] * rp[j];
    }
    oe[s] = occ[base + s] + sigm(ato) * tro;           // additive enhance
    pe[s] = prc[base + s] * (1.f + sigm(atp) * trp);   // multiplicative
  }
  float* hrow = hconv + (size_t)idx * SS;
#pragma unroll
  for (int t = 0; t < SS; ++t)
    hrow[t] = cW[0]*oe[t] + cW[1]*pe[t] + cW[2]*oe[t+1] + cW[3]*pe[t+1] + cb[0];
}

// ---------------------------------------------------------------------------
// 4a) qA[q][h] = relu(W0 @ I[l][q] + b0)   (same across batch)
// ---------------------------------------------------------------------------
__global__ __launch_bounds__(256)
void prep_qa_k(const float* __restrict__ I, const float* __restrict__ W,
               const float* __restrict__ bb, float* __restrict__ QA) {
  int idx = blockIdx.x * 256 + threadIdx.x;
  if (idx >= SET * HH) return;
  int q = idx / HH, h = idx % HH;
  float a = bb[h];
#pragma unroll
  for (int j = 0; j < HH; ++j) a += W[h * HH + j] * I[q * HH + j];
  QA[idx] = fmaxf(a, 0.f);
}

// 4b) k,v = relu(linear(X)) per node -> KV[(b*N+n)*2 + {0,1}][h]
__global__ __launch_bounds__(256)
void kv_a_k(const float* __restrict__ Xin, const float* __restrict__ W,
            const float* __restrict__ bb, float* __restrict__ KV) {
  int idx = blockIdx.x * 256 + threadIdx.x;
  if (idx >= NB * N_NODES) return;
  const float* Wk = W + 1 * 121;  const float* bk = bb + 1 * HH;
  const float* Wv = W + 2 * 121;  const float* bv = bb + 2 * HH;
  float x[HH];
  const float* xr = Xin + (size_t)idx * HH;
#pragma unroll
  for (int j = 0; j < HH; ++j) x[j] = xr[j];
  float* kout = KV + (size_t)idx * 2 * HH;
#pragma unroll
  for (int h = 0; h < HH; ++h) {
    float ak = bk[h], av = bv[h];
#pragma unroll
    for (int j = 0; j < HH; ++j) { ak += Wk[h*HH+j]*x[j]; av += Wv[h*HH+j]*x[j]; }
    kout[h]      = fmaxf(ak, 0.f);
    kout[HH + h] = fmaxf(av, 0.f);
  }
}

// 4c) MAB0 softmax over 8192 keys; one WG per (b,h,q). Writes aw0 + outA.
__device__ __forceinline__ float wave_max(float v) {
#pragma unroll
  for (int o = 16; o; o >>= 1) v = fmaxf(v, __shfl_xor(v, o, 32));
  return v;
}
__device__ __forceinline__ float wave_sum(float v) {
#pragma unroll
  for (int o = 16; o; o >>= 1) v += __shfl_xor(v, o, 32);
  return v;
}

__global__ __launch_bounds__(256)
void attn_a_k(const float* __restrict__ KV, const float* __restrict__ QA,
              float* __restrict__ aw0, float* __restrict__ outA) {
  __shared__ float r0[8], r1[8];
  const int r = blockIdx.x;            // ((b*11)+h)*32 + q
  const int q = r & 31, bh = r >> 5, h = bh % HH, b = bh / HH;
  const int wid = threadIdx.x >> 5, lane = threadIdx.x & 31;
  const float qv = QA[q * HH + h];

  float mx = -1e30f;
  for (int k = threadIdx.x; k < N_NODES; k += 256)
    mx = fmaxf(mx, qv * KV[((size_t)(b * N_NODES + k) * 2) * HH + h]);
  mx = wave_max(mx);
  if (lane == 0) r0[wid] = mx;
  __syncthreads();
  float gmax = r0[0];
#pragma unroll
  for (int i = 1; i < 8; ++i) gmax = fmaxf(gmax, r0[i]);
  __syncthreads();

  float se = 0.f, sv = 0.f;
  for (int k = threadIdx.x; k < N_NODES; k += 256) {
    const size_t kb = ((size_t)(b * N_NODES + k) * 2) * HH + h;
    const float e = __expf(qv * KV[kb] - gmax);
    se += e;  sv += e * KV[kb + HH];
  }
  se = wave_sum(se);  sv = wave_sum(sv);
  if (lane == 0) { r0[wid] = se; r1[wid] = sv; }
  __syncthreads();
  float tse = 0.f, tsv = 0.f;
#pragma unroll
  for (int i = 0; i < 8; ++i) { tse += r0[i]; tsv += r1[i]; }
  const float inv = 1.f / tse;

  float* dst = aw0 + ((size_t)(h * NB + b) * SET + q) * N_NODES;
  for (int k = threadIdx.x; k < N_NODES; k += 256) {
    const float e = __expf(qv * KV[((size_t)(b * N_NODES + k) * 2) * HH + h] - gmax);
    dst[k] = e * inv;                  // coalesced aw0 store
  }
  if (threadIdx.x == 0) outA[(b * SET + q) * HH + h] = tsv * inv;
}

// 4d) Hs = relu(W3 @ outA + b3), per (b,q)
__global__ __launch_bounds__(128)
void hs_k(const float* __restrict__ OA, const float* __restrict__ W,
          const float* __restrict__ bb, float* __restrict__ HS) {
  int idx = threadIdx.x;
  if (idx >= NB * SET) return;
  const float* W3 = W + 3 * 121;  const float* b3 = bb + 3 * HH;
  const float* in = OA + (size_t)idx * HH;
  float* out = HS + (size_t)idx * HH;
#pragma unroll
  for (int h = 0; h < HH; ++h) {
    float a = b3[h];
#pragma unroll
    for (int j = 0; j < HH; ++j) a += W3[h * HH + j] * in[j];
    out[h] = fmaxf(a, 0.f);
  }
}

// 4e) k2,v2 = relu(linear(Hs)) with MAB1 weights, per (b,k)
__global__ __launch_bounds__(128)
void kv2_k(const float* __restrict__ HS, const float* __restrict__ W,
           const float* __restrict__ bb, float* __restrict__ KV2) {
  int idx = threadIdx.x;
  if (idx >= NB * SET) return;
  const float* Wk = W + (4 + 1) * 121;  const float* bk = bb + 5 * HH;
  const float* Wv = W + (4 + 2) * 121;  const float* bv = bb + 6 * HH;
  const float* hr = HS + (size_t)idx * HH;
  float* out = KV2 + (size_t)idx * 2 * HH;
#pragma unroll
  for (int h = 0; h < HH; ++h) {
    float ak = bk[h], av = bv[h];
#pragma unroll
    for (int j = 0; j < HH; ++j) { ak += Wk[h*HH+j]*hr[j]; av += Wv[h*HH+j]*hr[j]; }
    out[h]      = fmaxf(ak, 0.f);
    out[HH + h] = fmaxf(av, 0.f);
  }
}

// 4f) MAB1 (32 keys, in registers) + output proj + fp linear. Writes aw1, z.
__global__ __launch_bounds__(256)
void attn_b_k(const float* __restrict__ Xin, const float* __restrict__ KV2,
              const float* __restrict__ W, const float* __restrict__ bb,
              const float* __restrict__ fpW, const float* __restrict__ fpb,
              float* __restrict__ aw1, float* __restrict__ Zout) {
  int idx = blockIdx.x * 256 + threadIdx.x;
  if (idx >= NB * N_NODES) return;
  const int b = idx >> 13, n = idx & (N_NODES - 1);
  const float* Wq = W + 4 * 121;        const float* bq = bb + 4 * HH;
  const float* Wo = W + (4 + 3) * 121;  const float* bo = bb + 7 * HH;
  float x[HH], q2[HH], o[HH];
  const float* xr = Xin + (size_t)idx * HH;
#pragma unroll
  for (int j = 0; j < HH; ++j) x[j] = xr[j];
#pragma unroll
  for (int h = 0; h < HH; ++h) {
    float a = bq[h];
#pragma unroll
    for (int j = 0; j < HH; ++j) a += Wq[h * HH + j] * x[j];
    q2[h] = fmaxf(a, 0.f);
  }
  for (int h = 0; h < HH; ++h) {
    float sc[SET];
    float mx = -1e30f;
#pragma unroll
    for (int k = 0; k < SET; ++k) {
      sc[k] = q2[h] * KV2[((size_t)(b * SET + k) * 2) * HH + h];
      mx = fmaxf(mx, sc[k]);
    }
    float se = 0.f;
#pragma unroll
    for (int k = 0; k < SET; ++k) { sc[k] = __expf(sc[k] - mx); se += sc[k]; }
    const float inv = 1.f / se;
    float acc = 0.f;
    float* dst = aw1 + ((size_t)(h * NB + b) * N_NODES + n) * SET;
#pragma unroll
    for (int k = 0; k < SET; ++k) {
      const float a = sc[k] * inv;
      dst[k] = a;
      acc += a * KV2[((size_t)(b * SET + k) * 2 + 1) * HH + h];
    }
    o[h] = acc;
  }
  float hp[HH];
#pragma unroll
  for (int h = 0; h < HH; ++h) {
    float a = bo[h];
#pragma unroll
    for (int j = 0; j < HH; ++j) a += Wo[h * HH + j] * o[j];
    hp[h] = fmaxf(a, 0.f);
  }
  float* z = Zout + (size_t)idx * SS;
#pragma unroll
  for (int s = 0; s < SS; ++s) {
    float a = fpb[s];
#pragma unroll
    for (int j = 0; j < HH; ++j) a += fpW[s * HH + j] * hp[j];
    z[s] = (a > 0.f) ? a : 0.01f * a;   // leaky_relu
  }
}

// ---------------------------------------------------------------------------
// 5) residual mix + 2-level TCN + FC head -> y (B,N)
// ---------------------------------------------------------------------------
__global__ __launch_bounds__(256)
void head_k(const float* __restrict__ HC, const float* __restrict__ Z1,
            const float* __restrict__ Z2, const float* __restrict__ tW,
            const float* __restrict__ tb, const float* __restrict__ f1W,
            const float* __restrict__ f1b, const float* __restrict__ f2W,
            const float* __restrict__ f2b, const float* __restrict__ f3W,
            const float* __restrict__ f3b, float* __restrict__ y) {
  int idx = blockIdx.x * 256 + threadIdx.x;
  if (idx >= NB * N_NODES) return;
  const float* hc = HC + (size_t)idx * SS;
  const float* z1 = Z1 + (size_t)idx * SS;
  const float* z2 = Z2 + (size_t)idx * SS;
  float xc[2][SS];
#pragma unroll
  for (int t = 0; t < SS; ++t) {
    const float inner = 0.5f * z1[t] + 0.5f * hc[t];
    const float outer = 0.5f * z2[t] + 0.5f * inner;
    xc[0][t] = outer;  xc[1][t] = outer;
  }
#pragma unroll
  for (int lvl = 0; lvl < 2; ++lvl) {
    const int dil = 1 << lvl;
    float o1[2][SS], o2[2][SS];
#pragma unroll
    for (int c = 0; c < 2; ++c)
      for (int t = 0; t < SS; ++t) {
        float a = tb[(lvl * 2 + 0) * 2 + c];
        for (int ci = 0; ci < 2; ++ci)
          for (int tap = 0; tap < 3; ++tap) {
            const int ts = t - (2 - tap) * dil;
            if (ts >= 0) a += tW[(((lvl*2+0)*2+c)*2+ci)*3+tap] * xc[ci][ts];
          }
        o1[c][t] = fmaxf(a, 0.f);
      }
#pragma unroll
    for (int c = 0; c < 2; ++c)
      for (int t = 0; t < SS; ++t) {
        float a = tb[(lvl * 2 + 1) * 2 + c];
        for (int ci = 0; ci < 2; ++ci)
          for (int tap = 0; tap < 3; ++tap) {
            const int ts = t - (2 - tap) * dil;
            if (ts >= 0) a += tW[(((lvl*2+1)*2+c)*2+ci)*3+tap] * o1[ci][ts];
          }
        o2[c][t] = fmaxf(a, 0.f);
      }
#pragma unroll
    for (int c = 0; c < 2; ++c)
      for (int t = 0; t < SS; ++t) xc[c][t] = fmaxf(o2[c][t] + xc[c][t], 0.f);
  }
  const float ht0 = xc[0][SS - 1], ht1 = xc[1][SS - 1];
  float Hcm[2][6], Hnm[2][2], av[2], vt[6];
#pragma unroll
  for (int c = 0; c < 2; ++c)
    for (int f = 0; f < 6; ++f) {
      float a = f1b[f];
      for (int t = 0; t < SS - 1; ++t) a += f1W[f * (SS - 1) + t] * xc[c][t];
      Hcm[c][f] = a;
    }
#pragma unroll
  for (int c = 0; c < 2; ++c)
    for (int j = 0; j < 2; ++j) {
      float a = f2b[j];
      for (int f = 0; f < 6; ++f) a += f2W[j * 6 + f] * Hcm[c][f];
      Hnm[c][j] = a;
    }
#pragma unroll
  for (int c = 0; c < 2; ++c)
    av[c] = sigm(Hnm[c][0] * ht0 + Hnm[c][1] * ht1);
#pragma unroll
  for (int f = 0; f < 6; ++f) vt[f] = av[0] * Hcm[0][f] + av[1] * Hcm[1][f];
  float yy = f3b[0];
#pragma unroll
  for (int f = 0; f < 6; ++f) yy += f3W[f] * vt[f];
  yy += f3W[6] * ht0 + f3W[7] * ht1;
  y[idx] = yy;
}

// ---------------------------------------------------------------------------
extern "C" void kernel_launch(void* const* d_in, const int* in_sizes, int n_in,
                              void* d_out, int out_size, void* d_ws, size_t ws_size,
                              hipStream_t stream) {
  const float* occ  = (const float*)d_in[0];
  const float* prc  = (const float*)d_in[1];
  const float* adj  = (const float*)d_in[2];
  const float* rwtW = (const float*)d_in[3];
  const float* rwtb = (const float*)d_in[4];
  const float* rwaW = (const float*)d_in[5];
  const float* rwab = (const float*)d_in[6];
  const float* cW   = (const float*)d_in[7];
  const float* cb   = (const float*)d_in[8];
  const float* baI  = (const float*)d_in[9];
  const float* baW  = (const float*)d_in[10];
  const float* bab  = (const float*)d_in[11];
  const float* fpW  = (const float*)d_in[12];
  const float* fpb  = (const float*)d_in[13];
  const float* tW   = (const float*)d_in[14];
  const float* tb   = (const float*)d_in[15];
  const float* f1W  = (const float*)d_in[16];
  const float* f1b  = (const float*)d_in[17];
  const float* f2W  = (const float*)d_in[18];
  const float* f2b  = (const float*)d_in[19];
  const float* f3W  = (const float*)d_in[20];
  const float* f3b  = (const float*)d_in[21];
  float* out = (float*)d_out;

  // workspace layout (floats)
  float* ws  = (float*)d_ws;
  float* Xp  = ws;                         // 786432 (fragment-swizzled X)
  float* RW  = Xp  + 786432;               // 786432
  float* HC  = RW  + 786432;               // 360448
  float* Z1  = HC  + 360448;               // 360448
  float* Z2  = Z1  + 360448;               // 360448
  float* KV  = Z2  + 360448;               // 720896
  float* QA  = KV  + 720896;               // 352
  float* OA  = QA  + 352;                  // 1408
  float* HS  = OA  + 1408;                 // 1408
  float* KV2 = HS  + 1408;                 // 2816

  const size_t AWSZ = (size_t)HH * NB * SET * N_NODES;   // 11,534,336

  pack_x_k<<<(N_NODES * NCOLS + 255) / 256, 256, 0, stream>>>(occ, prc, Xp);
  gemm_rw_k<<<N_NODES / 64, 128, 0, stream>>>(adj, Xp, RW);
  enhance_conv_k<<<(NB * N_NODES + 255) / 256, 256, 0, stream>>>(
      RW, occ, prc, rwtW, rwtb, rwaW, rwab, cW, cb, HC);

  for (int l = 0; l < 2; ++l) {
    const float* Xin = (l == 0) ? HC : Z1;
    float* Zout      = (l == 0) ? Z1 : Z2;
    float* aw0 = out + NB * N_NODES + (size_t)l * 2 * AWSZ;
    float* aw1 = aw0 + AWSZ;
    const float* Wl = baW + (size_t)l * 2 * 4 * 121;
    const float* bl = bab + (size_t)l * 2 * 4 * HH;
    prep_qa_k<<<2, 256, 0, stream>>>(baI + (size_t)l * SET * HH, Wl, bl, QA);
    kv_a_k<<<(NB * N_NODES + 255) / 256, 256, 0, stream>>>(Xin, Wl, bl, KV);
    attn_a_k<<<NB * HH * SET, 256, 0, stream>>>(KV, QA, aw0, OA);
    hs_k<<<1, 128, 0, stream>>>(OA, Wl, bl, HS);
    kv2_k<<<1, 128, 0, stream>>>(HS, Wl, bl, KV2);
    attn_b_k<<<(NB * N_NODES + 255) / 256, 256, 0, stream>>>(
        Xin, KV2, Wl, bl, fpW + (size_t)l * 121, fpb + (size_t)l * HH, aw1, Zout);
  }

  head_k<<<(NB * N_NODES + 255) / 256, 256, 0, stream>>>(
      HC, Z1, Z2, tW, tb, f1W, f1b, f2W, f2b, f3W, f3b, out);
}